// Linear_73237782331549
// MI455X (gfx1250) — compile-verified
//
#include <hip/hip_runtime.h>

#define BATCH 16384
#define NFEAT 26
#define VOCAB 1000000
#define NDENSE 13
#define BN_EPS 1e-5f

typedef __attribute__((ext_vector_type(2))) float v2f;
typedef __attribute__((ext_vector_type(8))) float v8f;

// ---------------------------------------------------------------------------
// Kernel 1: per-block partial sums / sum-of-squares for BatchNorm statistics.
// 64 blocks x 256 threads, one row per thread. Deterministic LDS tree reduce.
// Output: partial[block*26 + d]      = sum of column d over block's 256 rows
//         partial[block*26 + 13 + d] = sum of squares
// ---------------------------------------------------------------------------
__global__ __launch_bounds__(256) void bn_partial_kernel(
    const float* __restrict__ dense, float* __restrict__ partial) {
  __shared__ float ssum[256 * NDENSE];
  __shared__ float ssq [256 * NDENSE];
  const int t   = threadIdx.x;
  const int row = blockIdx.x * 256 + t;
#pragma unroll
  for (int d = 0; d < NDENSE; ++d) {
    float v = dense[row * NDENSE + d];
    ssum[t * NDENSE + d] = v;
    ssq [t * NDENSE + d] = v * v;
  }
  __syncthreads();
  for (int stride = 128; stride > 0; stride >>= 1) {
    if (t < stride) {
#pragma unroll
      for (int d = 0; d < NDENSE; ++d) {
        ssum[t * NDENSE + d] += ssum[(t + stride) * NDENSE + d];
        ssq [t * NDENSE + d] += ssq [(t + stride) * NDENSE + d];
      }
    }
    __syncthreads();
  }
  if (t < NDENSE) {
    partial[blockIdx.x * 26 + t]      = ssum[t];
    partial[blockIdx.x * 26 + 13 + t] = ssq[t];
  }
}

// ---------------------------------------------------------------------------
// Kernel 2: finalize BN stats and fold the whole dense affine chain into
//   coef[d] = gamma[d]*invstd[d]*weight[d]           (d = 0..12)
//   coef[13] = bias + sum_d (beta[d]-mean[d]*gamma[d]*invstd[d])*weight[d]
// so dense_logit[b] = sum_d dense[b,d]*coef[d] + coef[13].
// ---------------------------------------------------------------------------
__global__ void bn_finalize_kernel(
    const float* __restrict__ partial, const float* __restrict__ gamma,
    const float* __restrict__ beta, const float* __restrict__ weight,
    const float* __restrict__ bias, float* __restrict__ coef) {
  __shared__ float cshare[NDENSE];
  const int d = threadIdx.x;
  if (d < NDENSE) {
    float s = 0.f, q = 0.f;
    for (int i = 0; i < 64; ++i) {
      s += partial[i * 26 + d];
      q += partial[i * 26 + 13 + d];
    }
    float mean = s * (1.0f / BATCH);
    float var  = q * (1.0f / BATCH) - mean * mean;   // biased variance (jnp.var)
    float inv  = rsqrtf(var + BN_EPS);
    float g    = gamma[d] * inv;
    coef[d]    = g * weight[d];
    cshare[d]  = (beta[d] - mean * g) * weight[d];
  }
  __syncthreads();
  if (d == 0) {
    float c = bias[0];
#pragma unroll
    for (int i = 0; i < NDENSE; ++i) c += cshare[i];
    coef[13] = c;
  }
}

// ---------------------------------------------------------------------------
// Fetch one element of the virtual 16x40 per-tile A matrix:
//   k in [0,26)  : embedding gather  emb[k*V + ids[row*26+k]]
//   k in [26,39) : dense[row,k-26] * coef[k-26]
//   k == 39     : 0 (pad)
// ---------------------------------------------------------------------------
__device__ __forceinline__ float fetch_val(
    int row, int k, const int* __restrict__ ids,
    const float* __restrict__ dense, const float* __restrict__ emb,
    const float* __restrict__ coef) {
  float v = 0.0f;
  if (k < NFEAT) {
    int id = ids[row * NFEAT + k];
    v = emb[(size_t)k * (size_t)VOCAB + (size_t)(unsigned)id];
  } else if (k < NFEAT + NDENSE) {
    int d = k - NFEAT;
    v = dense[row * NDENSE + d] * coef[d];
  }
  return v;
}

// ---------------------------------------------------------------------------
// Kernel 3: fused gather + row-sum via WMMA.
// Each wave handles a 16-row tile. The 40-long per-row vector (26 emb values,
// 13 scaled dense values, 1 zero pad) is fed as A (16x4 chunks) into
// v_wmma_f32_16x16x4_f32 with B == all-ones, so D[m,0] accumulates the row
// sum. B = splat(1.0) is layout-independent. C/D f32 layout (ISA 7.12.2):
// VGPR r holds M=r on lanes 0-15 and M=8+r on lanes 16-31, N = lane%16, so
// lane 0 ends with rows base..base+7 and lane 16 with rows base+8..base+15
// in 8 consecutive VGPRs -> two float4 stores per lane.
// ---------------------------------------------------------------------------
__global__ __launch_bounds__(256) void fused_main_kernel(
    const int* __restrict__ ids, const float* __restrict__ dense,
    const float* __restrict__ emb, const float* __restrict__ coef,
    float* __restrict__ out) {
  const int lane = threadIdx.x & 31;
  const int wave = threadIdx.x >> 5;
  const int base = (blockIdx.x * 8 + wave) * 16;   // 16 rows per wave
  const int hi   = lane >> 4;                      // 0 -> K pair (4c,4c+1); 1 -> (4c+2,4c+3)
  const int m    = lane & 15;                      // row within tile (A: M = lane%16)
  const int row  = base + m;

  const float c0 = coef[NDENSE];                   // additive constant

  v8f acc = {0.f, 0.f, 0.f, 0.f, 0.f, 0.f, 0.f, 0.f};
  v2f ones;
  ones.x = 1.0f;
  ones.y = 1.0f;

#pragma unroll
  for (int c = 0; c < 10; ++c) {                   // 10 chunks of K=4 cover 40 values
    const int k0 = 4 * c + 2 * hi;
    v2f a;
    a.x = fetch_val(row, k0,     ids, dense, emb, coef);
    a.y = fetch_val(row, k0 + 1, ids, dense, emb, coef);
    // D = A x ones + C  ==> per-row partial sums accumulate in every column.
    acc = __builtin_amdgcn_wmma_f32_16x16x4_f32(
        /*neg_a=*/false, a, /*neg_b=*/false, ones,
        /*c_mod=*/(short)0, acc, /*reuse_a=*/false, /*reuse_b=*/false);
  }

  // N==0 column lives on lanes 0 (M = 0..7) and 16 (M = 8..15).
  if (m == 0) {
    float4 lo, hi4;
    lo.x  = acc[0] + c0; lo.y  = acc[1] + c0; lo.z  = acc[2] + c0; lo.w  = acc[3] + c0;
    hi4.x = acc[4] + c0; hi4.y = acc[5] + c0; hi4.z = acc[6] + c0; hi4.w = acc[7] + c0;
    float* p = out + base + hi * 8;                // 32B-aligned
    *(float4*)(p)     = lo;
    *(float4*)(p + 4) = hi4;
  }
}

// ---------------------------------------------------------------------------
// Launch wrapper
// ---------------------------------------------------------------------------
extern "C" void kernel_launch(void* const* d_in, const int* in_sizes, int n_in,
                              void* d_out, int out_size, void* d_ws, size_t ws_size,
                              hipStream_t stream) {
  const int*   ids    = (const int*)d_in[0];    // [B,F]
  const float* dense  = (const float*)d_in[1];  // [B,D]
  const float* emb    = (const float*)d_in[2];  // [F,V]
  const float* gamma  = (const float*)d_in[3];  // [D]
  const float* beta   = (const float*)d_in[4];  // [D]
  const float* weight = (const float*)d_in[5];  // [D,1]
  const float* bias   = (const float*)d_in[6];  // [1]
  float*       out    = (float*)d_out;          // [B,1]

  float* wsf     = (float*)d_ws;
  float* coef    = wsf;                         // 14 floats (+pad)
  float* partial = wsf + 16;                    // 64 * 26 floats

  bn_partial_kernel<<<64, 256, 0, stream>>>(dense, partial);
  bn_finalize_kernel<<<1, 32, 0, stream>>>(partial, gamma, beta, weight, bias, coef);
  fused_main_kernel<<<BATCH / 128, 256, 0, stream>>>(ids, dense, emb, coef, out);
}